// Attention_7121055776934
// MI455X (gfx1250) — compile-verified
//
#include <hip/hip_runtime.h>
#include <cstdint>

// TransformerXL relative attention for MI455X (gfx1250, wave32, WMMA + TDM).
// Matrix math: V_WMMA_F32_16X16X4_F32 (exact fp32, matches fp32 reference).
// LDS staging: TENSOR_LOAD_TO_LDS (Tensor Data Mover) with LDS row padding
// done by the DMA engine (pad_interval/pad_amount), completion via TENSORcnt.
//
// Fragment layouts (ISA 7.12.2, 32-bit, wave32):
//   A (16x4):  lane L holds row m=L&15; VGPR j holds K = j + 2*(L>>4)
//   B (4x16):  lane L holds col n=L&15; VGPR j holds K = j + 2*(L>>4)
//   C/D(16x16):VGPR i holds row i + 8*(L>>4), col L&15

typedef float v2f __attribute__((ext_vector_type(2)));
typedef float v8f __attribute__((ext_vector_type(8)));
typedef unsigned int u32x4 __attribute__((ext_vector_type(4)));
typedef int i32x8 __attribute__((ext_vector_type(8)));
typedef int i32x4 __attribute__((ext_vector_type(4)));

#define NEG_INF_F (-1e9f)

constexpr int Bsz = 4;
constexpr int Qn  = 512;
constexpr int Mn  = 512;
constexpr int Rn  = 1024;   // Q + M
constexpr int Dn  = 1024;
constexpr int Hn  = 16;
constexpr int Sn  = 64;     // D / H

__device__ __forceinline__ v8f vzero8() {
  v8f v;
#pragma unroll
  for (int i = 0; i < 8; ++i) v[i] = 0.f;
  return v;
}

__device__ __forceinline__ v8f wmma_f32(v2f a, v2f b, v8f c) {
  // (neg_a, A, neg_b, B, c_mod, C, reuse_a, reuse_b)
  return __builtin_amdgcn_wmma_f32_16x16x4_f32(false, a, false, b, (short)0, c,
                                               false, false);
}

// LDS byte offset of a __shared__ object (generic LDS pointers carry the LDS
// offset in their low 32 bits on AMDGPU).
__device__ __forceinline__ uint32_t lds_off(const void* p) {
  return (uint32_t)(uintptr_t)p;
}

// ---------------------------------------------------------------------------
// Tensor Data Mover: 2D tile (global -> LDS), fp32 elements.
//   td0/td1      tensor dims (elements/rows) for hardware OOB zero-fill
//   tile0/tile1  tile dims (elements per row / rows)
//   stride0      row stride in elements
//   pad_int      pad interval code: pads after 2^(pad_int+1) DWORDs
//   pad_amt      pad amount code: (pad_amt+1) DWORDs inserted into LDS addr
// D# packing per CDNA5 ISA (group0) / §8.4 (group1).
// This toolchain's builtin is the 6-arg clang-23 form:
//   (u32x4 g0, i32x8 g1, i32x4 g2, i32x4 g3, i32x8 extra, i32 cpol)
// ---------------------------------------------------------------------------
__device__ __forceinline__ void tdm_load_2d(uint32_t lds_byte, const float* g,
                                            uint32_t td0, uint32_t td1,
                                            uint32_t tile0, uint32_t tile1,
                                            uint32_t stride0, uint32_t pad_int,
                                            uint32_t pad_amt) {
  const uint64_t ga = (uint64_t)(uintptr_t)g;
  u32x4 g0;
  g0.x = 1u;                                  // count=1, user descriptor
  g0.y = lds_byte;                            // lds_addr [63:32]
  g0.z = (uint32_t)ga;                        // global_addr [95:64]
  g0.w = (uint32_t)(ga >> 32) | (2u << 30);   // global_addr hi | type=2
  i32x8 g1;
  g1[0] = (int)((2u << 16) |                  // data_size = 4 bytes
                (1u << 20) |                  // pad_enable
                (pad_int << 22) | (pad_amt << 25));
  g1[1] = (int)((td0 & 0xFFFFu) << 16);       // atomic_barrier=0 | dim0 lo16
  g1[2] = (int)(((td0 >> 16) & 0xFFFFu) | ((td1 & 0xFFFFu) << 16));
  g1[3] = (int)(((td1 >> 16) & 0xFFFFu) | ((tile0 & 0xFFFFu) << 16));
  g1[4] = (int)(tile1 & 0xFFFFu);             // tile_dim1 | tile_dim2=0
  g1[5] = (int)stride0;                       // dim0_stride [191:160]
  g1[6] = 0;                                  // dim0_stride hi | dim1_stride lo
  g1[7] = 0;                                  // dim1_stride hi
  i32x4 z4;
  z4[0] = z4[1] = z4[2] = z4[3] = 0;          // groups 2/3 unused (2D)
  i32x8 z8;
#pragma unroll
  for (int i = 0; i < 8; ++i) z8[i] = 0;
  __builtin_amdgcn_tensor_load_to_lds(g0, g1, z4, z4, z8, 0);
}

// ---------------------------------------------------------------------------
// Generic 64x64-tile GEMM: C[row,n] = sum_k A(row,k) * W[k*1024+n]
// MODE 0: A = query_seqs [B*Q,1024]; out -> [B,H,Q,S]
// MODE 1: A = concat(memory,query) rows; out -> [B,H,R,S]
//         (M=512 is 64-aligned, so a row tile never straddles the concat seam)
// MODE 2: A = positional_encoding [R,1024]; out -> [H,R,S]
// MODE 3: A = attn output [B,H,Q,S] regathered as [B*Q,(h,s)]; out -> [B*Q,1024]
//         (k-blocks of 32 stay inside one head -> plain 2D tile, stride S)
// ---------------------------------------------------------------------------
template <int MODE>
__device__ __forceinline__ void storeC(float* __restrict__ out, int row, int n,
                                       float v) {
  if constexpr (MODE == 0) {
    int b = row >> 9, t = row & (Qn - 1), h = n >> 6, s = n & 63;
    out[((b * Hn + h) * Qn + t) * Sn + s] = v;
  } else if constexpr (MODE == 1) {
    int b = row >> 10, t = row & (Rn - 1), h = n >> 6, s = n & 63;
    out[((b * Hn + h) * Rn + t) * Sn + s] = v;
  } else if constexpr (MODE == 2) {
    int h = n >> 6, s = n & 63;
    out[(h * Rn + row) * Sn + s] = v;
  } else {  // MODE 3
    out[row * Dn + n] = v;
  }
}

template <int MODE>
__global__ __launch_bounds__(128) void gemm64(const float* __restrict__ A0,
                                              const float* __restrict__ A1,
                                              const float* __restrict__ W,
                                              float* __restrict__ out) {
  __shared__ float As[64][36];   // 64 rows x 32 K  (TDM pads 4 DW per 32 DW)
  __shared__ float Wls[32][68];  // 32 K x 64 cols (TDM pads 4 DW per 64 DW)

  const int tid = threadIdx.x;
  const int lane = tid & 31, wid = tid >> 5;
  const int half = lane >> 4, l15 = lane & 15;
  const int rowBase = blockIdx.y * 64;
  const int colBase = blockIdx.x * 64;

  v8f acc[4];
#pragma unroll
  for (int t = 0; t < 4; ++t) acc[t] = vzero8();

  for (int k0 = 0; k0 < Dn; k0 += 32) {
    __syncthreads();
    if (wid == 0) {
      const float* abase;
      uint32_t rstride;
      if constexpr (MODE == 0 || MODE == 2) {
        abase = A0 + (size_t)rowBase * Dn + k0;
        rstride = Dn;
      } else if constexpr (MODE == 1) {
        int b = rowBase >> 10, t = rowBase & (Rn - 1);
        abase = (t < Mn) ? A0 + (size_t)(b * Mn + t) * Dn + k0
                         : A1 + (size_t)(b * Qn + (t - Mn)) * Dn + k0;
        rstride = Dn;
      } else {  // MODE 3
        int b = rowBase >> 9, t = rowBase & (Qn - 1);
        int hh = k0 >> 6, s = k0 & 63;
        abase = A0 + (size_t)((b * Hn + hh) * Qn + t) * Sn + s;
        rstride = Sn;
      }
      tdm_load_2d(lds_off(&As[0][0]), abase, 32, 64, 32, 64, rstride, 4, 3);
      tdm_load_2d(lds_off(&Wls[0][0]), W + (size_t)k0 * Dn + colBase, 64, 32,
                  64, 32, Dn, 5, 3);
      __builtin_amdgcn_s_wait_tensorcnt(0);
    }
    __syncthreads();
#pragma unroll
    for (int kk = 0; kk < 32; kk += 4) {
      const int ka = kk + 2 * half;
      v2f a;
      a.x = As[wid * 16 + l15][ka];
      a.y = As[wid * 16 + l15][ka + 1];
#pragma unroll
      for (int t = 0; t < 4; ++t) {
        v2f b;
        b.x = Wls[ka][t * 16 + l15];
        b.y = Wls[ka + 1][t * 16 + l15];
        acc[t] = wmma_f32(a, b, acc[t]);
      }
    }
  }
#pragma unroll
  for (int t = 0; t < 4; ++t)
#pragma unroll
    for (int i = 0; i < 8; ++i)
      storeC<MODE>(out, rowBase + wid * 16 + i + 8 * half,
                   colBase + t * 16 + l15, acc[t][i]);
}

// ---------------------------------------------------------------------------
// Fused attention with exact TransformerXL rel-shift + online softmax.
// Grid: (Q/64, B*H). Block: 128 threads (4 waves x 16 q-rows).
// shifted[q,r]: r<=M+q  -> praw[q, r+Q-1-q]            (window W1)
//               r==M+q+1-> 0
//               else    -> praw[q+1, r-q-M-2]          (window W2)
// Both read the window GEMM at skew x = (r-r0) + 63 - (q-q0), x in [0,127).
// TDM OOB zero-fill (tensor_dim) replaces all manual row clamps.
// ---------------------------------------------------------------------------
__global__ __launch_bounds__(128) void attn_fused(
    const float* __restrict__ qz, const float* __restrict__ kz,
    const float* __restrict__ vz, const float* __restrict__ relz,
    const float* __restrict__ cbias, const float* __restrict__ pbias,
    const float* __restrict__ mask, float* __restrict__ oz) {
  __shared__ float qs[65][68];     // q rows q0..q0+64 (row 64 zero-filled past Q)
  __shared__ float ks[64][68];     // K chunk  [key][s]
  __shared__ float vs[64][68];     // V chunk  [key][s]
  __shared__ float rels[128][68];  // rel window [x][s]
  __shared__ float wsm[64][132];   // W1/W2 results, later P (cols 0..63)
  __shared__ float cb[64];
  __shared__ float pb[64];

  const int tid = threadIdx.x;
  const int lane = tid & 31, wid = tid >> 5;
  const int half = lane >> 4, l15 = lane & 15;
  const int bh = blockIdx.y;  // b*H + h
  const int h = bh & (Hn - 1);
  const int q0 = blockIdx.x * 64;

  if (wid == 0) {
    // q tile: 65 rows, hardware zero-fills rows past Q.
    tdm_load_2d(lds_off(&qs[0][0]), qz + (size_t)(bh * Qn + q0) * Sn, Sn,
                (uint32_t)(Qn - q0), Sn, 65, Sn, 5, 3);
  }
  if (tid < 64)
    cb[tid] = cbias[h * Sn + tid];
  else
    pb[tid - 64] = pbias[h * Sn + (tid - 64)];

  float mrow[8], lrow[8];
  v8f oacc[4];
#pragma unroll
  for (int i = 0; i < 8; ++i) {
    mrow[i] = -1e30f;
    lrow[i] = 0.f;
  }
#pragma unroll
  for (int t = 0; t < 4; ++t) oacc[t] = vzero8();

#pragma unroll 1
  for (int rc = 0; rc < Rn / 64; ++rc) {
    const int r0 = rc * 64;
    const bool anyNoWrap = (r0 <= Mn + q0 + 63);  // some r <= M+q in tile
    const bool anyWrap = (r0 + 63 > Mn + q0);     // some r >  M+q in tile
    const int j0 = r0 + Qn - 64 - q0;             // W1 window base (>=0)

    __syncthreads();
    if (wid == 0) {
      tdm_load_2d(lds_off(&ks[0][0]), kz + (size_t)(bh * Rn + r0) * Sn, Sn, 64,
                  Sn, 64, Sn, 5, 3);
      tdm_load_2d(lds_off(&vs[0][0]), vz + (size_t)(bh * Rn + r0) * Sn, Sn, 64,
                  Sn, 64, Sn, 5, 3);
      if (anyNoWrap) {
        // under anyNoWrap, j0 <= 1023; rows past R zero-fill in hardware
        tdm_load_2d(lds_off(&rels[0][0]), relz + (size_t)(h * Rn + j0) * Sn,
                    Sn, (uint32_t)(Rn - j0), Sn, 128, Sn, 5, 3);
      }
      __builtin_amdgcn_s_wait_tensorcnt(0);
    }
    __syncthreads();

    float sc[4][8];
    {  // content = (q + cbias) . K^T
      v8f cacc[4];
#pragma unroll
      for (int t = 0; t < 4; ++t) cacc[t] = vzero8();
#pragma unroll
      for (int kk = 0; kk < 64; kk += 4) {
        const int ka = kk + 2 * half;
        v2f a;
        a.x = qs[wid * 16 + l15][ka] + cb[ka];
        a.y = qs[wid * 16 + l15][ka + 1] + cb[ka + 1];
#pragma unroll
        for (int t = 0; t < 4; ++t) {
          v2f b;
          b.x = ks[t * 16 + l15][ka];
          b.y = ks[t * 16 + l15][ka + 1];
          cacc[t] = wmma_f32(a, b, cacc[t]);
        }
      }
#pragma unroll
      for (int t = 0; t < 4; ++t)
#pragma unroll
        for (int i = 0; i < 8; ++i) sc[t][i] = cacc[t][i];
    }

    if (anyNoWrap) {  // W1 window, A rows q0..q0+63
#pragma unroll 1
      for (int wt = 0; wt < 8; ++wt) {
        v8f wacc = vzero8();
#pragma unroll
        for (int kk = 0; kk < 64; kk += 4) {
          const int ka = kk + 2 * half;
          v2f a, b;
          a.x = qs[wid * 16 + l15][ka] + pb[ka];
          a.y = qs[wid * 16 + l15][ka + 1] + pb[ka + 1];
          b.x = rels[wt * 16 + l15][ka];
          b.y = rels[wt * 16 + l15][ka + 1];
          wacc = wmma_f32(a, b, wacc);
        }
#pragma unroll
        for (int i = 0; i < 8; ++i)
          wsm[wid * 16 + i + 8 * half][wt * 16 + l15] = wacc[i];
      }
#pragma unroll
      for (int t = 0; t < 4; ++t)
#pragma unroll
        for (int i = 0; i < 8; ++i) {
          const int qm = wid * 16 + i + 8 * half;
          const int col = t * 16 + l15;
          if (r0 + col <= Mn + q0 + qm) sc[t][i] += wsm[qm][col + 63 - qm];
        }
    }

    if (anyWrap) {  // W2 window, A rows q0+1..q0+64
      __syncthreads();
      const int j20 = r0 - q0 - Mn - 65;
      const int xs = (j20 < 0) ? -j20 : 0;  // xs < 128 whenever anyWrap
      if (wid == 0) {
        // rows x < xs are never consumed (cond requires j2 = j20+x >= 0)
        tdm_load_2d(lds_off(&rels[0][0]) + (uint32_t)xs * 68u * 4u,
                    relz + (size_t)(h * Rn + j20 + xs) * Sn, Sn,
                    (uint32_t)(Rn - (j20 + xs)), Sn, (uint32_t)(128 - xs), Sn,
                    5, 3);
        __builtin_amdgcn_s_wait_tensorcnt(0);
      }
      __syncthreads();
#pragma unroll 1
      for (int wt = 0; wt < 8; ++wt) {
        v8f wacc = vzero8();
#pragma unroll
        for (int kk = 0; kk < 64; kk += 4) {
          const int ka = kk + 2 * half;
          v2f a, b;
          a.x = qs[wid * 16 + l15 + 1][ka] + pb[ka];
          a.y = qs[wid * 16 + l15 + 1][ka + 1] + pb[ka + 1];
          b.x = rels[wt * 16 + l15][ka];
          b.y = rels[wt * 16 + l15][ka + 1];
          wacc = wmma_f32(a, b, wacc);
        }
#pragma unroll
        for (int i = 0; i < 8; ++i)
          wsm[wid * 16 + i + 8 * half][wt * 16 + l15] = wacc[i];
      }
#pragma unroll
      for (int t = 0; t < 4; ++t)
#pragma unroll
        for (int i = 0; i < 8; ++i) {
          const int qm = wid * 16 + i + 8 * half;
          const int col = t * 16 + l15;
          if (r0 + col >= Mn + q0 + qm + 2) sc[t][i] += wsm[qm][col + 63 - qm];
        }
    }

    // scale 1/sqrt(S) and token mask
#pragma unroll
    for (int t = 0; t < 4; ++t)
#pragma unroll
      for (int i = 0; i < 8; ++i) {
        const int qm = wid * 16 + i + 8 * half;
        const int col = t * 16 + l15;
        sc[t][i] =
            sc[t][i] * 0.125f + mask[(q0 + qm) * Rn + r0 + col] * NEG_INF_F;
      }

    // online softmax over this chunk (row reductions in 16-lane halves)
#pragma unroll
    for (int i = 0; i < 8; ++i) {
      float mx = fmaxf(fmaxf(sc[0][i], sc[1][i]), fmaxf(sc[2][i], sc[3][i]));
#pragma unroll
      for (int d = 1; d < 16; d <<= 1) mx = fmaxf(mx, __shfl_xor(mx, d, 32));
      const float mnew = fmaxf(mrow[i], mx);
      const float f = __expf(mrow[i] - mnew);
      mrow[i] = mnew;
      float rs = 0.f;
#pragma unroll
      for (int t = 0; t < 4; ++t) {
        const float p = __expf(sc[t][i] - mnew);
        sc[t][i] = p;
        rs += p;
      }
#pragma unroll
      for (int d = 1; d < 16; d <<= 1) rs += __shfl_xor(rs, d, 32);
      lrow[i] = lrow[i] * f + rs;
#pragma unroll
      for (int ts = 0; ts < 4; ++ts) oacc[ts][i] = oacc[ts][i] * f;
    }

    // stage P (D-layout -> A-layout) through this wave's wsm rows
#pragma unroll
    for (int t = 0; t < 4; ++t)
#pragma unroll
      for (int i = 0; i < 8; ++i)
        wsm[wid * 16 + i + 8 * half][t * 16 + l15] = sc[t][i];

    // O += P * V
#pragma unroll
    for (int kk = 0; kk < 64; kk += 4) {
      const int ka = kk + 2 * half;
      v2f a;
      a.x = wsm[wid * 16 + l15][ka];
      a.y = wsm[wid * 16 + l15][ka + 1];
#pragma unroll
      for (int ts = 0; ts < 4; ++ts) {
        v2f b;
        b.x = vs[ka][ts * 16 + l15];
        b.y = vs[ka + 1][ts * 16 + l15];
        oacc[ts] = wmma_f32(a, b, oacc[ts]);
      }
    }
  }

  // normalize and write [B,H,Q,S]
#pragma unroll
  for (int ts = 0; ts < 4; ++ts)
#pragma unroll
    for (int i = 0; i < 8; ++i) {
      const int qm = wid * 16 + i + 8 * half;
      oz[(bh * Qn + q0 + qm) * Sn + ts * 16 + l15] = oacc[ts][i] / lrow[i];
    }
}

// ---------------------------------------------------------------------------
extern "C" void kernel_launch(void* const* d_in, const int* in_sizes, int n_in,
                              void* d_out, int out_size, void* d_ws,
                              size_t ws_size, hipStream_t stream) {
  (void)in_sizes;
  (void)n_in;
  (void)out_size;
  (void)ws_size;
  const float* query = (const float*)d_in[0];   // [B,Q,D]
  const float* pos   = (const float*)d_in[1];   // [R,D]
  const float* tmask = (const float*)d_in[2];   // [1,1,Q,R]
  const float* mem   = (const float*)d_in[3];   // [B,M,D]
  const float* w_q   = (const float*)d_in[4];   // [D,H,S]
  const float* w_k   = (const float*)d_in[5];
  const float* w_v   = (const float*)d_in[6];
  const float* w_r   = (const float*)d_in[7];
  const float* w_o   = (const float*)d_in[8];   // [H,S,D]
  const float* cbias = (const float*)d_in[9];   // [H,S]
  const float* pbias = (const float*)d_in[10];  // [H,S]

  float* ws = (float*)d_ws;
  float* qbuf = ws;                                 // [B,H,Q,S]  2M floats
  float* kbuf = qbuf + (size_t)Bsz * Hn * Qn * Sn;  // [B,H,R,S]  4M
  float* vbuf = kbuf + (size_t)Bsz * Hn * Rn * Sn;  // [B,H,R,S]  4M
  float* rbuf = vbuf + (size_t)Bsz * Hn * Rn * Sn;  // [H,R,S]    1M
  float* obuf = rbuf + (size_t)Hn * Rn * Sn;        // [B,H,Q,S]  2M

  // projections (64x64 tiles, 128 threads / 4 waves)
  gemm64<0><<<dim3(16, (Bsz * Qn) / 64), 128, 0, stream>>>(query, nullptr, w_q,
                                                           qbuf);
  gemm64<1><<<dim3(16, (Bsz * Rn) / 64), 128, 0, stream>>>(mem, query, w_k,
                                                           kbuf);
  gemm64<1><<<dim3(16, (Bsz * Rn) / 64), 128, 0, stream>>>(mem, query, w_v,
                                                           vbuf);
  gemm64<2><<<dim3(16, Rn / 64), 128, 0, stream>>>(pos, nullptr, w_r, rbuf);

  // fused rel-attention
  attn_fused<<<dim3(Qn / 64, Bsz * Hn), 128, 0, stream>>>(
      qbuf, kbuf, vbuf, rbuf, cbias, pbias, tmask, obuf);

  // output projection -> d_out [B,Q,D]
  gemm64<3><<<dim3(16, (Bsz * Qn) / 64), 128, 0, stream>>>(obuf, nullptr, w_o,
                                                           (float*)d_out);
}